// GCN_49417893708128
// MI455X (gfx1250) — compile-verified
//
#include <hip/hip_runtime.h>
#include <math.h>

// Problem constants (match reference)
#define NN   50000   // nodes
#define EE   800000  // edges
#define IN_  128
#define HH   256     // H == O == 256
#define GG   64      // graphs
#define CC   64      // classes

typedef float v2f __attribute__((ext_vector_type(2)));
typedef float v8f __attribute__((ext_vector_type(8)));

// ---------------- degree / norm ----------------
__global__ void deg_kernel(const int* __restrict__ dst, float* __restrict__ deg, int E) {
    int e = blockIdx.x * blockDim.x + threadIdx.x;
    if (e < E) atomicAdd(&deg[dst[e]], 1.0f);
}

__global__ void dinv_kernel(const float* __restrict__ deg, float* __restrict__ dinv, int n) {
    int i = blockIdx.x * blockDim.x + threadIdx.x;
    if (i < n) dinv[i] = rsqrtf(deg[i] + 1.0f);   // self-loop included
}

// ---------------- fp32 WMMA GEMM: C[M,Nf] = A[M,K] @ B[K,Nf] ----------------
// One wave computes one 16x16 tile via V_WMMA_F32_16X16X4_F32.
// A frag (16x4):  lane<16 -> {K=k0,k0+1}, lane>=16 -> {K=k0+2,k0+3}, M = lane&15
// B frag (4x16):  same K split per lane-half, N = lane&15
// C/D (16x16 f32, 8 VGPRs): row = r + 8*(lane>>4), col = lane&15
__global__ void gemm_wmma_f32(const float* __restrict__ A,
                              const float* __restrict__ B,
                              float* __restrict__ C,
                              int M, int K, int Nf) {
    const int wave  = threadIdx.x >> 5;
    const int lane  = threadIdx.x & 31;
    const int tid   = blockIdx.x * (blockDim.x >> 5) + wave;
    const int tilesN = Nf >> 4;
    const int tm = tid / tilesN;
    const int tn = tid % tilesN;
    if (tm * 16 >= M) return;            // wave-uniform; EXEC stays all-ones for WMMA

    const int half = lane >> 4;          // 0: K pair {0,1}, 1: K pair {2,3}
    const int l15  = lane & 15;
    const int row  = tm * 16 + l15;      // M index for A frag
    const int col  = tn * 16 + l15;      // N index for B frag / C

    v8f acc = {0.f, 0.f, 0.f, 0.f, 0.f, 0.f, 0.f, 0.f};
    for (int k0 = 0; k0 < K; k0 += 4) {
        const int kk = k0 + 2 * half;
        v2f a, b;
        a.x = A[(size_t)row * K + kk];
        a.y = A[(size_t)row * K + kk + 1];
        b.x = B[(size_t)kk * Nf + col];
        b.y = B[(size_t)(kk + 1) * Nf + col];
        acc = __builtin_amdgcn_wmma_f32_16x16x4_f32(
            /*neg_a=*/false, a, /*neg_b=*/false, b,
            /*c_mod=*/(short)0, acc, /*reuse_a=*/false, /*reuse_b=*/false);
    }
#pragma unroll
    for (int r = 0; r < 8; ++r) {
        const int orow = tm * 16 + r + 8 * half;
        C[(size_t)orow * Nf + col] = acc[r];
    }
}

// ---------------- edge scatter: agg[dst] += t[src] * dinv[src]*dinv[dst] ----------------
// One wave per edge: 32 lanes x 8 floats = 256 features. L2-resident atomics.
__global__ void scatter_kernel(const float* __restrict__ t,
                               const int* __restrict__ src,
                               const int* __restrict__ dst,
                               const float* __restrict__ dinv,
                               float* __restrict__ agg, int E) {
    const int gid  = blockIdx.x * blockDim.x + threadIdx.x;
    const int e    = gid >> 5;
    const int lane = gid & 31;
    if (e >= E) return;
    const int s = src[e];
    const int d = dst[e];
    const float w = dinv[s] * dinv[d];
    const float4* ts = (const float4*)(t + (size_t)s * HH);
    float* ad = agg + (size_t)d * HH;
    const float4 v0 = ts[lane * 2 + 0];
    const float4 v1 = ts[lane * 2 + 1];
    const int f = lane * 8;
    atomicAdd(ad + f + 0, v0.x * w);
    atomicAdd(ad + f + 1, v0.y * w);
    atomicAdd(ad + f + 2, v0.z * w);
    atomicAdd(ad + f + 3, v0.w * w);
    atomicAdd(ad + f + 4, v1.x * w);
    atomicAdd(ad + f + 5, v1.y * w);
    atomicAdd(ad + f + 6, v1.z * w);
    atomicAdd(ad + f + 7, v1.w * w);
}

// ---------------- combine: t = [gelu](agg + t*dinv^2 + bias) (in place on t) ----------------
__global__ void combine_kernel(float* __restrict__ t,
                               const float* __restrict__ agg,
                               const float* __restrict__ dinv,
                               const float* __restrict__ bias,
                               int n, int applyGelu) {
    const int idx = blockIdx.x * blockDim.x + threadIdx.x;
    if (idx >= n) return;
    const int i = idx >> 8;    // node (HH == 256)
    const int f = idx & 255;   // feature
    float sn = dinv[i];
    sn = sn * sn;              // self-loop norm 1/deg
    float v = agg[idx] + t[idx] * sn + bias[f];
    if (applyGelu) v = 0.5f * v * (1.0f + erff(v * 0.70710678118654752440f)); // exact GELU
    t[idx] = v;
}

// ---------------- mean pool ----------------
__global__ void pool_kernel(const float* __restrict__ h, const int* __restrict__ batch,
                            float* __restrict__ pooled, int n) {
    const int idx = blockIdx.x * blockDim.x + threadIdx.x;
    if (idx >= n) return;
    const int i = idx >> 8;
    const int f = idx & 255;
    atomicAdd(&pooled[(size_t)batch[i] * HH + f], h[idx]);
}

__global__ void cnt_kernel(const int* __restrict__ batch, float* __restrict__ cnt, int n) {
    const int i = blockIdx.x * blockDim.x + threadIdx.x;
    if (i < n) atomicAdd(&cnt[batch[i]], 1.0f);
}

// ---------------- head: out[g,c] = (pooled[g,:]/cnt[g]) @ Wfc + bfc ----------------
__global__ void final_kernel(const float* __restrict__ pooled, const float* __restrict__ cnt,
                             const float* __restrict__ Wfc, const float* __restrict__ bfc,
                             float* __restrict__ out) {
    const int idx = blockIdx.x * blockDim.x + threadIdx.x;
    if (idx >= GG * CC) return;
    const int g = idx / CC;
    const int c = idx % CC;
    const float inv = 1.0f / fmaxf(cnt[g], 1.0f);
    float s = 0.f;
#pragma unroll 4
    for (int k = 0; k < HH; ++k)
        s += pooled[(size_t)g * HH + k] * Wfc[(size_t)k * CC + c];
    out[idx] = s * inv + bfc[c];
}

extern "C" void kernel_launch(void* const* d_in, const int* in_sizes, int n_in,
                              void* d_out, int out_size, void* d_ws, size_t ws_size,
                              hipStream_t stream) {
    const float* x     = (const float*)d_in[0];
    const int*   ei    = (const int*)  d_in[1];
    const int*   batch = (const int*)  d_in[2];
    const float* W1    = (const float*)d_in[3];
    const float* b1    = (const float*)d_in[4];
    const float* W2    = (const float*)d_in[5];
    const float* b2    = (const float*)d_in[6];
    const float* Wfc   = (const float*)d_in[7];
    const float* bfc   = (const float*)d_in[8];
    const int* src = ei;        // edge_index[0]
    const int* dst = ei + EE;   // edge_index[1]

    // workspace layout (floats)
    float* bufA   = (float*)d_ws;                 // N*HH  (t1 -> h1 -> agg2)
    float* bufB   = bufA + (size_t)NN * HH;       // N*HH  (agg1 -> t2 -> h2)
    float* deg    = bufB + (size_t)NN * HH;       // N
    float* dinv   = deg + NN;                     // N
    float* pooled = dinv + NN;                    // G*HH
    float* cnt    = pooled + (size_t)GG * HH;     // G

    const size_t matBytes = (size_t)NN * HH * sizeof(float);
    const int gemmBlocks  = (NN / 16) * (HH / 16) / 8;   // 8 waves (16x16 tiles) per block

    // degrees + symmetric norm
    hipMemsetAsync(deg, 0, NN * sizeof(float), stream);
    deg_kernel<<<(EE + 255) / 256, 256, 0, stream>>>(dst, deg, EE);
    dinv_kernel<<<(NN + 255) / 256, 256, 0, stream>>>(deg, dinv, NN);

    // layer 1: t1 = x @ W1 ; agg1 = scatter(t1) ; h1 = gelu(agg1 + t1/deg + b1)
    gemm_wmma_f32<<<gemmBlocks, 256, 0, stream>>>(x, W1, bufA, NN, IN_, HH);
    hipMemsetAsync(bufB, 0, matBytes, stream);
    scatter_kernel<<<(EE * 32) / 256, 256, 0, stream>>>(bufA, src, dst, dinv, bufB, EE);
    combine_kernel<<<(NN * HH) / 256, 256, 0, stream>>>(bufA, bufB, dinv, b1, NN * HH, 1);

    // layer 2: t2 = h1 @ W2 ; agg2 = scatter(t2) ; h2 = agg2 + t2/deg + b2
    gemm_wmma_f32<<<gemmBlocks, 256, 0, stream>>>(bufA, W2, bufB, NN, HH, HH);
    hipMemsetAsync(bufA, 0, matBytes, stream);
    scatter_kernel<<<(EE * 32) / 256, 256, 0, stream>>>(bufB, src, dst, dinv, bufA, EE);
    combine_kernel<<<(NN * HH) / 256, 256, 0, stream>>>(bufB, bufA, dinv, b2, NN * HH, 0);

    // mean pool + head
    hipMemsetAsync(pooled, 0, (size_t)GG * HH * sizeof(float), stream);
    hipMemsetAsync(cnt, 0, GG * sizeof(float), stream);
    pool_kernel<<<(NN * HH) / 256, 256, 0, stream>>>(bufB, batch, pooled, NN * HH);
    cnt_kernel<<<(NN + 255) / 256, 256, 0, stream>>>(batch, cnt, NN);
    final_kernel<<<(GG * CC + 255) / 256, 256, 0, stream>>>(pooled, cnt, Wfc, bfc, (float*)d_out);
}